// FFTMagAgc_60825326846433
// MI455X (gfx1250) — compile-verified
//
#include <hip/hip_runtime.h>
#include <hip/hip_bf16.h>

typedef __attribute__((ext_vector_type(2))) float v2f;
typedef __attribute__((ext_vector_type(4))) float v4f;
typedef __attribute__((ext_vector_type(8))) float v8f;

#define B_ 48
#define T_ 5000
#define F_ 161
#define FP 164   // bins padded to multiple of 4 (zero-filled)
#define NK (FP / 4)  // 41 K-chunks of the 16x16x4 WMMA

// ---------------------------------------------------------------------------
// Pass 1: per-frame weighted power sum via V_WMMA_F32_16X16X4_F32.
// Block = 128 threads (4 waves); each wave owns 16 consecutive frames.
// LDS holds weighted mag^2 (w=2 interior, w=1 for DC/Nyquist, 0 padding).
// A-tile (16x4 f32): lane l -> M = l%16, K = 2*(l/16) + {0,1}  (ISA layout).
// B = all-ones 4x16 -> D[m][n] = rowsum(A[m]); read column N=0:
//   lane 0  holds M=0..7 in c[0..7], lane 16 holds M=8..15 in c[0..7].
// ---------------------------------------------------------------------------
__global__ __launch_bounds__(128) void k_framepow(const float* __restrict__ mag,
                                                  float* __restrict__ xout) {
    __shared__ float lds[4 * 16 * FP];
    const int lane = threadIdx.x & 31;
    const int wave = threadIdx.x >> 5;
    const long frame0 = (long)blockIdx.x * 64 + (long)wave * 16;
    float* wl = &lds[wave * 16 * FP];

    // Stage weighted spec_pow into LDS, coalesced across the wave.
    for (int r = 0; r < 16; ++r) {
        const float* row = mag + (frame0 + r) * F_;
        for (int f = lane; f < FP; f += 32) {
            float w = (f == 0 || f == F_ - 1) ? 1.0f : 2.0f;
            float v = (f < F_) ? row[f] : 0.0f;
            wl[r * FP + f] = w * v * v;
        }
    }
    __syncthreads();

    const int m    = lane & 15;
    const int koff = (lane >> 4) << 1;       // 0 for lanes 0-15, 2 for 16-31
    const float* arow = wl + m * FP + koff;

    v8f c = {};
    const v2f bones = {1.0f, 1.0f};          // all-ones B regardless of lane
    #pragma unroll 4
    for (int k = 0; k < NK; ++k) {
        v2f a;
        a.x = arow[4 * k];                   // K = koff   (VGPR0)
        a.y = arow[4 * k + 1];               // K = koff+1 (VGPR1)
        c = __builtin_amdgcn_wmma_f32_16x16x4_f32(
                /*neg_a=*/false, a, /*neg_b=*/false, bones,
                /*c_mod=*/(short)0, c, /*reuse_a=*/false, /*reuse_b=*/false);
    }

    const float scale = 1.0f / 320.0f;
    if (lane == 0) {
        #pragma unroll
        for (int r = 0; r < 8; ++r) xout[frame0 + r] = sqrtf(c[r] * scale);
    } else if (lane == 16) {
        #pragma unroll
        for (int r = 0; r < 8; ++r) xout[frame0 + 8 + r] = sqrtf(c[r] * scale);
    }
}

// ---------------------------------------------------------------------------
// Pass 2: EMA scan per batch row. g[n] = 0.9 g[n-1] + 0.1 x[n], g[0] = x[0].
// Linear-recurrence composite (A,B): g_out = A*g_in + B; Kogge-Stone over a
// wave32 chunk via __shfl_up, sequential carry across 157 chunks.
// Also emits inv = 1/(g + 0.001) so pass 3 is multiply-only.
// ---------------------------------------------------------------------------
__global__ __launch_bounds__(32) void k_scan(const float* __restrict__ x,
                                             float* __restrict__ gain_out,
                                             float* __restrict__ inv_out) {
    const int b = blockIdx.x;
    const int lane = threadIdx.x;
    const float* xr = x + (long)b * T_;
    float* gr = gain_out + (long)b * T_;
    float* ir = inv_out + (long)b * T_;

    float carry = 0.0f;
    for (int t0 = 0; t0 < T_; t0 += 32) {
        const int t = t0 + lane;
        const float xv = (t < T_) ? xr[t] : 0.0f;
        float A  = (t == 0) ? 0.0f : 0.9f;          // frame 0: passthrough
        float Bv = (t == 0) ? xv   : 0.1f * xv;
        #pragma unroll
        for (int d = 1; d < 32; d <<= 1) {
            float a2 = __shfl_up(A, d);
            float b2 = __shfl_up(Bv, d);
            if (lane >= d) { Bv = A * b2 + Bv; A = A * a2; }
        }
        const float g = A * carry + Bv;
        if (t < T_) {
            gr[t] = g;
            ir[t] = 1.0f / (g + 0.001f);
        }
        carry = __shfl(g, 31);
    }
}

// ---------------------------------------------------------------------------
// Pass 3: new_mag = mag * inv[frame].  float4 vectorized (38,640,000 % 4 == 0).
// mag should still be L2-resident (154.6 MB < 192 MB L2) from pass 1.
// Nontemporal stores: new_mag is never re-read.
// ---------------------------------------------------------------------------
__global__ __launch_bounds__(256) void k_norm(const float* __restrict__ mag,
                                              const float* __restrict__ inv,
                                              float* __restrict__ out) {
    const unsigned i4 = ((unsigned)blockIdx.x * 256u + threadIdx.x) * 4u;
    const unsigned N = (unsigned)B_ * T_ * F_;
    if (i4 >= N) return;
    const unsigned f0 = i4 / F_;                 // frame of first element
    const int r = (int)(i4 - f0 * F_);
    const float inv0 = inv[f0];
    const float inv1 = (r + 3 >= F_) ? inv[f0 + 1] : inv0;
    const v4f m4 = *(const v4f*)(mag + i4);
    v4f o;
    o.x = m4.x * inv0;                           // r < 161 always
    o.y = m4.y * ((r + 1 >= F_) ? inv1 : inv0);
    o.z = m4.z * ((r + 2 >= F_) ? inv1 : inv0);
    o.w = m4.w * ((r + 3 >= F_) ? inv1 : inv0);
    __builtin_nontemporal_store(o, (v4f*)(out + i4));
}

extern "C" void kernel_launch(void* const* d_in, const int* in_sizes, int n_in,
                              void* d_out, int out_size, void* d_ws, size_t ws_size,
                              hipStream_t stream) {
    (void)in_sizes; (void)n_in; (void)out_size; (void)ws_size;
    const float* mag = (const float*)d_in[0];
    float* out    = (float*)d_out;
    float* newmag = out;                               // 48*5000*161 floats
    float* gain   = out + (size_t)B_ * T_ * F_;        // 48*5000 floats
    float* xbuf   = (float*)d_ws;                      // 240,000 floats
    float* invbuf = xbuf + (size_t)B_ * T_;            // 240,000 floats

    // Pass 1: 240,000 frames, 64 frames/block -> 3750 blocks
    k_framepow<<<(B_ * T_) / 64, 128, 0, stream>>>(mag, xbuf);
    // Pass 2: one wave per batch row
    k_scan<<<B_, 32, 0, stream>>>(xbuf, gain, invbuf);
    // Pass 3: 9,660,000 float4s
    const int n4 = (B_ * T_ * F_) / 4;
    k_norm<<<(n4 + 255) / 256, 256, 0, stream>>>(mag, invbuf, newmag);
}